// BrainRNN_438086664755
// MI455X (gfx1250) — compile-verified
//
#include <hip/hip_runtime.h>
#include <math.h>

#define DD    1024
#define NN    8192
#define BATCH 32
#define KCHUNK 512

typedef float v2f __attribute__((ext_vector_type(2)));
typedef float v8f __attribute__((ext_vector_type(8)));

// byte offset of a shared-memory object inside the LDS aperture
__device__ __forceinline__ unsigned lds_off(const void* p) {
    return (unsigned)(unsigned long long)(const __attribute__((address_space(3))) void*)p;
}

// CDNA5 async copy: 16 bytes per lane, global -> LDS, tracked by ASYNCcnt
__device__ __forceinline__ void async_copy_b128(unsigned lds, const void* gaddr) {
    asm volatile("global_load_async_to_lds_b128 %0, %1, off"
                 :: "v"(lds), "v"((unsigned long long)gaddr)
                 : "memory");
}

__device__ __forceinline__ void wait_async0() {
    asm volatile("s_wait_asynccnt 0x0" ::: "memory");
}

// t[b*ld + d] = bias[d]
__global__ void init_bias_kernel(float* __restrict__ t, const float* __restrict__ bias, int ld, int n) {
    int i = blockIdx.x * blockDim.x + threadIdx.x;
    if (i < n) t[i] = bias[i % ld];
}

__global__ void sigmoid_kernel(const float* __restrict__ t, float* __restrict__ xx, int n) {
    int i = blockIdx.x * blockDim.x + threadIdx.x;
    if (i < n) xx[i] = 1.0f / (1.0f + __expf(-t[i]));
}

// Masked skinny GEMM with split-K atomics:
//   t[b, d] += sum_{c in [c_begin,c_end)} in[b*ldin + c] * W[d*ldw + c] * M(d,c)
//   M(d,c) = adj[(rowoff + c)*NN + coloff + d]   (if use_mask)
// Block tile: 64 d-columns (blockIdx.x*64), all 32 batch rows, KCHUNK-deep K slab (blockIdx.y).
__launch_bounds__(128)
__global__ void masked_gemm_kernel(const float* __restrict__ in,
                                   const float* __restrict__ W,
                                   const int*   __restrict__ adj,
                                   float*       __restrict__ t,
                                   int ldin, int ldw, int ldout,
                                   int c_begin, int c_end,
                                   int rowoff, int coloff,
                                   int use_mask)
{
    // double-buffered tiles; row stride 36 floats (144B): 16B aligned, no 16-row bank clash
    __shared__ float sIn[2][32][36];   // [b][c]
    __shared__ float sW [2][64][36];   // [d][c]
    __shared__ int   sAdj[2][32][64];  // [c][d]

    const int tid  = threadIdx.x;
    const int d0   = blockIdx.x * 64;
    const int cb   = c_begin + blockIdx.y * KCHUNK;
    const int ce   = min(cb + KCHUNK, c_end);

    const int wave = tid >> 5;
    const int lane = tid & 31;
    const int half = lane >> 4;   // 0: lanes 0-15, 1: lanes 16-31
    const int l16  = lane & 15;
    const int dl   = wave * 16;   // wave's local d sub-tile

    // issue all async copies for one 32-deep K chunk into buffer bf
    auto issue_stage = [&](int bf, int c) {
        #pragma unroll
        for (int j = 0; j < 2; ++j) {            // activations: 32 rows x 128B
            int e = tid + j * 128;
            int b = e >> 3, seg = (e & 7) * 4;
            async_copy_b128(lds_off(&sIn[bf][b][seg]),
                            in + (size_t)b * ldin + c + seg);
        }
        #pragma unroll
        for (int j = 0; j < 4; ++j) {            // weights: 64 rows x 128B
            int e = tid + j * 128;
            int d = e >> 3, seg = (e & 7) * 4;
            async_copy_b128(lds_off(&sW[bf][d][seg]),
                            W + (size_t)(d0 + d) * ldw + c + seg);
        }
        if (use_mask) {
            #pragma unroll
            for (int j = 0; j < 4; ++j) {        // adj: 32 rows x 256B
                int e = tid + j * 128;
                int r = e >> 4, seg = (e & 15) * 4;
                async_copy_b128(lds_off(&sAdj[bf][r][seg]),
                                adj + (size_t)(rowoff + c + r) * NN + coloff + d0 + seg);
            }
        }
    };

    v8f cLo = {};  // batch rows 0..15
    v8f cHi = {};  // batch rows 16..31

    int buf = 0;
    issue_stage(0, cb);

    for (int c = cb; c < ce; c += 32) {
        wait_async0();          // my async copies for `buf` are in LDS
        __syncthreads();        // everyone's copies done; prev compute on buf^1 done
        if (c + 32 < ce) issue_stage(buf ^ 1, c + 32);

        // 8 x K=4 WMMA steps over this 32-deep chunk
        #pragma unroll
        for (int kk = 0; kk < 32; kk += 4) {
            const int ks = kk + 2 * half;  // VGPR j of A/B holds K = j + 2*half
            v2f a_lo = *(const v2f*)&sIn[buf][l16][ks];
            v2f a_hi = *(const v2f*)&sIn[buf][16 + l16][ks];
            v2f bb   = *(const v2f*)&sW[buf][dl + l16][ks];
            if (use_mask) {
                bb.x *= (float)sAdj[buf][ks][dl + l16];
                bb.y *= (float)sAdj[buf][ks + 1][dl + l16];
            }
            cLo = __builtin_amdgcn_wmma_f32_16x16x4_f32(false, a_lo, false, bb, (short)0, cLo, false, false);
            cHi = __builtin_amdgcn_wmma_f32_16x16x4_f32(false, a_hi, false, bb, (short)0, cHi, false, false);
        }
        buf ^= 1;
    }

    // C/D layout: VGPR r -> M = r + 8*half, N = l16
    const int col = d0 + dl + l16;
    #pragma unroll
    for (int r = 0; r < 8; ++r) {
        int b = r + 8 * half;
        atomicAdd(&t[(size_t)b        * ldout + col], cLo[r]);
        atomicAdd(&t[(size_t)(16 + b) * ldout + col], cHi[r]);
    }
}

extern "C" void kernel_launch(void* const* d_in, const int* in_sizes, int n_in,
                              void* d_out, int out_size, void* d_ws, size_t ws_size,
                              hipStream_t stream) {
    const float* x    = (const float*)d_in[0];   // 32 x 256
    const float* h    = (const float*)d_in[1];   // 32 x 8192
    const int*   adj  = (const int*)  d_in[2];   // 8192 x 8192
    const float* W_in = (const float*)d_in[3];   // 1024 x 256
    const float* b_in = (const float*)d_in[4];   // 1024
    const float* W_h  = (const float*)d_in[5];   // 7 x 1024 x 1024
    const float* b_h  = (const float*)d_in[6];   // 7 x 1024
    const float* W_r  = (const float*)d_in[7];   // 7 x 1024 x 8192
    const float* W_s  = (const float*)d_in[8];   // 6 x 1024 x 6144
    const float* W_o  = (const float*)d_in[9];   // 64 x 1024
    const float* b_o  = (const float*)d_in[10];  // 64
    float* out = (float*)d_out;                  // 32 x 64

    float* t     = (float*)d_ws;          // 32*1024 accumulator
    float* skips = t + BATCH * DD;        // 8 buffers of 32*1024

    auto gemm = [&](const float* in_p, const float* W_p, int ldin, int ldw,
                    int ldout, float* out_p, int c0, int c1,
                    int rowoff, int coloff, int use_mask) {
        dim3 grid(ldout / 64, (c1 - c0 + KCHUNK - 1) / KCHUNK);
        masked_gemm_kernel<<<grid, 128, 0, stream>>>(in_p, W_p, adj, out_p,
                                                     ldin, ldw, ldout, c0, c1,
                                                     rowoff, coloff, use_mask);
    };
    const int nT = BATCH * DD;

    // ---- layer 0: xx = sigmoid(x @ W_in.T + b_in + h @ Wr_m(0).T) ----
    init_bias_kernel<<<(nT + 255) / 256, 256, 0, stream>>>(t, b_in, DD, nT);
    gemm(x, W_in, 256, 256, DD, t, 0, 256, 0, 0, 0);
    // Wr_m(0): mask adj[c, 0*D + d], rows c < 1*D zeroed -> c in [D, N)
    gemm(h, W_r, NN, NN, DD, t, DD, NN, 0, 0, 1);
    sigmoid_kernel<<<(nT + 255) / 256, 256, 0, stream>>>(t, skips, nT);

    // ---- layers i = 0..6 ----
    for (int i = 0; i < 7; ++i) {
        init_bias_kernel<<<(nT + 255) / 256, 256, 0, stream>>>(t, b_h + (size_t)i * DD, DD, nT);
        // Wh_m(i): mask adj[i*D + c, (i+1)*D + d]
        gemm(skips + (size_t)i * nT, W_h + (size_t)i * DD * DD, DD, DD, DD, t,
             0, DD, i * DD, (i + 1) * DD, 1);
        if (i >= 1 && i <= 5) {
            // + h @ Wr_m(i+1).T : mask adj[c, (i+1)D + d], c in [(i+2)D, N)
            gemm(h, W_r + (size_t)(i + 1) * DD * NN, NN, NN, DD, t,
                 (i + 2) * DD, NN, 0, (i + 1) * DD, 1);
        }
        if (i == 0) {
            // + h @ Wr_m(1).T : mask adj[c, D + d], c in [2D, N)
            gemm(h, W_r + (size_t)1 * DD * NN, NN, NN, DD, t, 2 * DD, NN, 0, DD, 1);
        } else {
            // + sc @ Ws_m(i-1).T, sc = concat(skips[0..i-1]):
            // decompose into per-segment GEMMs: segment j contributes
            //   skips[j] @ (W_s[i-1][:, jD:(j+1)D] * adj[jD + c, (i+1)D + d]).T
            for (int j = 0; j < i; ++j) {
                gemm(skips + (size_t)j * nT,
                     W_s + (size_t)(i - 1) * DD * 6 * DD + (size_t)j * DD,
                     DD, 6 * DD, DD, t,
                     0, DD, j * DD, (i + 1) * DD, 1);
            }
        }
        sigmoid_kernel<<<(nT + 255) / 256, 256, 0, stream>>>(t, skips + (size_t)(i + 1) * nT, nT);
    }

    // ---- output: out = xx @ W_o.T + b_o ----
    init_bias_kernel<<<(BATCH * 64 + 255) / 256, 256, 0, stream>>>(out, b_o, 64, BATCH * 64);
    gemm(skips + (size_t)7 * nT, W_o, DD, DD, 64, out, 0, DD, 0, 0, 0);
}